// ControlStep_41162966565016
// MI455X (gfx1250) — compile-verified
//
#include <hip/hip_runtime.h>
#include <hip/hip_bf16.h>

#define B_ 64
#define S_ 4096
#define H_ 256

typedef __attribute__((ext_vector_type(2))) float v2f;
typedef __attribute__((ext_vector_type(8))) float v8f;

// ---------------------------------------------------------------------------
// Kernel 1a: one-pass online-softmax attention pooling (partial).
// grid = B_*split blocks, 256 threads (8 waves). Block g handles batch
// b = g/split, row chunk g%split of size S_/split. Each wave owns 8 h-channels
// per lane (h = lane*8..lane*8+7) and an online (m, l, acc[8]) state.
// Masked rows (att == 0 exactly in the reference) are skipped -> ~2x less HBM.
// ---------------------------------------------------------------------------
__global__ __launch_bounds__(256) void attnpool_partial(
    const float* __restrict__ inp, const float* __restrict__ mask,
    const float* __restrict__ query, int split,
    float* __restrict__ ws_m, float* __restrict__ ws_l,
    float* __restrict__ ws_acc)
{
    const int g     = blockIdx.x;
    const int b     = g / split;
    const int chunk = g % split;
    const int tid   = threadIdx.x;
    const int lane  = tid & 31;
    const int wave  = tid >> 5;               // 0..7
    const int hbase = lane * 8;               // this lane's 8 channels

    const int rowsPerBlock = S_ / split;
    const int base = chunk * rowsPerBlock;

    __shared__ float s_m[8];
    __shared__ float s_l[8];
    __shared__ float s_acc[8][H_];

    // preload this lane's slice of the query
    const float* q = query + b * H_ + hbase;
    float4 q0 = *(const float4*)(q);
    float4 q1 = *(const float4*)(q + 4);

    const float* rowbase = inp + (size_t)b * S_ * H_;
    const float* mrow    = mask + (size_t)b * S_;

    float m = -3.0e38f;
    float l = 0.0f;
    float acc[8];
#pragma unroll
    for (int j = 0; j < 8; ++j) acc[j] = 0.0f;

    const int iters = rowsPerBlock >> 3;      // rows per wave
    for (int i = 0; i < iters; ++i) {
        const int s = base + wave + (i << 3);

        // prefetch 4 of this wave's rows ahead, only if that row is unmasked
        const int sp = s + 32;
        if (sp < base + rowsPerBlock && mrow[sp] != 0.0f)
            __builtin_prefetch(rowbase + (size_t)sp * H_ + hbase, 0, 1);

        const float mk = mrow[s];             // wave-uniform
        if (mk != 0.0f) {
            const float* r = rowbase + (size_t)s * H_ + hbase;
            float4 x0 = *(const float4*)(r);
            float4 x1 = *(const float4*)(r + 4);

            float d = x0.x * q0.x + x0.y * q0.y + x0.z * q0.z + x0.w * q0.w
                    + x1.x * q1.x + x1.y * q1.y + x1.z * q1.z + x1.w * q1.w;
            // wave32 butterfly reduction -> full dot product in every lane
#pragma unroll
            for (int off = 16; off >= 1; off >>= 1)
                d += __shfl_xor(d, off, 32);

            const float mn   = fmaxf(m, d);
            const float corr = __expf(m - mn);
            const float p    = __expf(d - mn);
            l = l * corr + p;
            acc[0] = acc[0] * corr + p * x0.x;
            acc[1] = acc[1] * corr + p * x0.y;
            acc[2] = acc[2] * corr + p * x0.z;
            acc[3] = acc[3] * corr + p * x0.w;
            acc[4] = acc[4] * corr + p * x1.x;
            acc[5] = acc[5] * corr + p * x1.y;
            acc[6] = acc[6] * corr + p * x1.z;
            acc[7] = acc[7] * corr + p * x1.w;
            m = mn;
        }
    }

    // intra-block combine of the 8 wave states
    if (lane == 0) { s_m[wave] = m; s_l[wave] = l; }
#pragma unroll
    for (int j = 0; j < 8; ++j) s_acc[wave][hbase + j] = acc[j];
    __syncthreads();

    // thread tid owns channel h = tid
    float M = s_m[0];
#pragma unroll
    for (int w = 1; w < 8; ++w) M = fmaxf(M, s_m[w]);
    float L = 0.0f, v = 0.0f;
#pragma unroll
    for (int w = 0; w < 8; ++w) {
        const float sw = __expf(s_m[w] - M);
        L += s_l[w] * sw;
        v += s_acc[w][tid] * sw;
    }
    ws_acc[(size_t)g * H_ + tid] = v;
    if (tid == 0) { ws_m[g] = M; ws_l[g] = L; }
}

// ---------------------------------------------------------------------------
// Kernel 1b: combine per-chunk partial softmax states -> extracted_msg.
// grid = B_ blocks, 256 threads; thread tid owns channel h = tid.
// ---------------------------------------------------------------------------
__global__ __launch_bounds__(256) void attnpool_combine(
    const float* __restrict__ ws_m, const float* __restrict__ ws_l,
    const float* __restrict__ ws_acc, int split, float* __restrict__ ext)
{
    const int b = blockIdx.x;
    const int h = threadIdx.x;

    float M = -3.0e38f;
    for (int c = 0; c < split; ++c) M = fmaxf(M, ws_m[b * split + c]);
    float L = 0.0f, v = 0.0f;
    for (int c = 0; c < split; ++c) {
        const int g = b * split + c;
        const float sw = __expf(ws_m[g] - M);
        L += ws_l[g] * sw;
        v += ws_acc[(size_t)g * H_ + h] * sw;
    }
    ext[b * H_ + h] = v / L;
}

// ---------------------------------------------------------------------------
// Kernel 2: control_emb = [query | extracted] @ W^T + bias via WMMA f32.
// out[m,n] = sum_k conc[m,k] * W[n,k],  M=64, K=512, N=256.
// One wave per 16x16 output tile: grid = (64/16)*(256/16) = 64 blocks of 32.
// Fragment layouts per CDNA5 ISA 7.12.2:
//   A 16x4 f32 : lanes 0-15 -> K={k0,k0+1}, lanes 16-31 -> K={k0+2,k0+3}
//   B 4x16 f32 : v0: lanes0-15 K=k0, lanes16-31 K=k0+2; v1: K=k0+1 / k0+3
//   C/D 16x16  : vr: lanes0-15 M=r (N=lane), lanes16-31 M=r+8 (N=lane-16)
// ---------------------------------------------------------------------------
__global__ __launch_bounds__(32) void linear_wmma(
    const float* __restrict__ query, const float* __restrict__ ext,
    const float* __restrict__ W, const float* __restrict__ bias,
    float* __restrict__ out)
{
    const int lane = threadIdx.x & 31;
    const int mt   = blockIdx.x >> 4;         // 0..3
    const int nt   = blockIdx.x & 15;         // 0..15
    const int m0   = mt << 4;
    const int n0   = nt << 4;
    const int half = lane >> 4;               // 0 or 1
    const int l15  = lane & 15;
    const int mrow = m0 + l15;
    const int ncol = n0 + l15;

    const float* qrow = query + (size_t)mrow * H_;
    const float* erow = ext   + (size_t)mrow * H_;
    const float* wrow = W     + (size_t)ncol * (2 * H_);

    v8f c = {};
#pragma unroll 4
    for (int k0 = 0; k0 < 2 * H_; k0 += 4) {
        const int kb = k0 + half * 2;         // kb even -> 8B aligned
        // A fragment: conc[mrow, kb..kb+1] (kb and kb+1 in the same half)
        v2f a = (kb < H_) ? *(const v2f*)(qrow + kb)
                          : *(const v2f*)(erow + (kb - H_));
        // B fragment: W[ncol, kb..kb+1]
        v2f bm = *(const v2f*)(wrow + kb);
        c = __builtin_amdgcn_wmma_f32_16x16x4_f32(
                /*neg_a=*/false, a, /*neg_b=*/false, bm,
                /*c_mod=*/(short)0, c, /*reuse_a=*/false, /*reuse_b=*/false);
    }

    const float bb = bias[ncol];
#pragma unroll
    for (int r = 0; r < 8; ++r) {
        const int mm = m0 + r + (half << 3);
        out[(size_t)mm * H_ + n0 + l15] = c[r] + bb;
    }
}

// ---------------------------------------------------------------------------
extern "C" void kernel_launch(void* const* d_in, const int* in_sizes, int n_in,
                              void* d_out, int out_size, void* d_ws, size_t ws_size,
                              hipStream_t stream) {
    const float* inp   = (const float*)d_in[0];  // [B,S,H]
    const float* mask  = (const float*)d_in[1];  // [B,S]
    const float* query = (const float*)d_in[2];  // [B,H]
    const float* W     = (const float*)d_in[3];  // [H,2H]
    const float* bias  = (const float*)d_in[4];  // [H]

    float* ext  = (float*)d_out;                 // output 0: [B,H]
    float* ctrl = ext + B_ * H_;                 // output 1: [B,H]

    // pick the largest split whose partial state fits in the workspace
    int split = 8;
    while (split > 1 &&
           (size_t)(B_ * split) * (size_t)(H_ + 2) * sizeof(float) > ws_size)
        split >>= 1;

    float* ws_m   = (float*)d_ws;
    float* ws_l   = ws_m + B_ * split;
    float* ws_acc = ws_l + B_ * split;

    attnpool_partial<<<B_ * split, 256, 0, stream>>>(inp, mask, query, split,
                                                     ws_m, ws_l, ws_acc);
    attnpool_combine<<<B_, 256, 0, stream>>>(ws_m, ws_l, ws_acc, split, ext);
    linear_wmma<<<(B_ / 16) * (H_ / 16), 32, 0, stream>>>(query, ext, W, bias, ctrl);
}